// MultiHeadAttention_1915555414301
// MI455X (gfx1250) — compile-verified
//
#include <hip/hip_runtime.h>

typedef float v2f __attribute__((ext_vector_type(2)));
typedef float v8f __attribute__((ext_vector_type(8)));

#define Bsz   4
#define Msz   2048
#define Hsz   16
#define Dsz   128
#define SCALEF 0.35355339059327373f   // (128//16)^-0.5 = 8^-0.5
#define LN_EPS 1e-5f

// workspace layout (float offsets)
#define WS_G    0                         // 16x16 = wq . wk^T
#define WS_GVF  256                       // 16x16 = wv . w_final^T
#define WS_MX   512                       // [64][2048] per-column max
#define WS_VD   (512 + 64*2048)           // [64][2048] Vc[jm]/den[j]
#define WS_AO   (512 + 2*64*2048)         // [4][2048] attention output (pre-LN)

// ---------------------------------------------------------------------------
// Kernel A: two 16x16x128 Gram matrices via chained V_WMMA_F32_16X16X4_F32.
// A-matrix 16x4 layout: lanes 0-15 hold M=lane, VGPR0/1 = K0/K1; lanes 16-31
// hold K2/K3.  B-matrix 4x16: lane%16 = N (column-major per lane), same K split.
// ---------------------------------------------------------------------------
__global__ __launch_bounds__(32)
void gram_wmma_kernel(const float* __restrict__ wq, const float* __restrict__ wk,
                      const float* __restrict__ wv, const float* __restrict__ wf,
                      float* __restrict__ ws) {
  const int lane = threadIdx.x;
  const int r    = lane & 15;           // M for A, N for B
  const int ko   = (lane >> 4) << 1;    // K sub-offset: 0 or 2
  v8f g  = {};
  v8f gv = {};
#pragma unroll
  for (int kk = 0; kk < 32; ++kk) {
    const int kb = kk * 4 + ko;
    v2f a, b, c, d;
    a.x = wq[r * 128 + kb];  a.y = wq[r * 128 + kb + 1];
    b.x = wk[r * 128 + kb];  b.y = wk[r * 128 + kb + 1];
    g = __builtin_amdgcn_wmma_f32_16x16x4_f32(false, a, false, b, (short)0, g,
                                              false, false);
    c.x = wv[r * 128 + kb];  c.y = wv[r * 128 + kb + 1];
    d.x = wf[r * 128 + kb];  d.y = wf[r * 128 + kb + 1];
    gv = __builtin_amdgcn_wmma_f32_16x16x4_f32(false, c, false, d, (short)0, gv,
                                               false, false);
  }
  // D layout: lane L holds N=L%16, rows M = vgpr + 8*(L>=16)
  const int mb = (lane >> 4) * 8;
#pragma unroll
  for (int rr = 0; rr < 8; ++rr) {
    ws[WS_G   + (mb + rr) * 16 + r] = g[rr];
    ws[WS_GVF + (mb + rr) * 16 + r] = gv[rr];
  }
}

// ---------------------------------------------------------------------------
// Kernel B (pass 1): per column j=(jm,jq): exact column max of
// S[i,j] = alpha * Qm[im] * G[iq,jq], alpha = SCALE*Kn[jm], then
// den[j] = sum_i exp(S - mx).  Stores mx[j] and Vc[jm]/den[j].
// grid (64, 8) x 256: one thread per column.
// ---------------------------------------------------------------------------
__global__ __launch_bounds__(256)
void pass1_kernel(const float* __restrict__ q, const float* __restrict__ k,
                  const float* __restrict__ v, float* __restrict__ ws) {
  __shared__ float sQ[128];
  __shared__ float sG[256];
  __shared__ float sBmax[16], sBmin[16];
  __shared__ float sQext[2];
  const int bh = blockIdx.x;
  const int b = bh >> 4, h = bh & 15;
  const int tid = threadIdx.x;

  if (tid < 128) sQ[tid] = q[b * Msz + h * 128 + tid];
  sG[tid] = ws[WS_G + tid];
  __syncthreads();

  if (tid == 0) {
    float mx = sQ[0], mn = sQ[0];
    for (int i = 1; i < 128; ++i) { float x = sQ[i]; mx = fmaxf(mx, x); mn = fminf(mn, x); }
    sQext[0] = mx; sQext[1] = mn;
  }
  __syncthreads();
  if (tid < 16) {   // per-jq extremes of Qm[im]*G[iq,jq] over (im,iq) -- exact
    const float qmx = sQext[0], qmn = sQext[1];
    float bmax = -INFINITY, bmin = INFINITY;
    for (int iq = 0; iq < 16; ++iq) {
      const float gg = sG[iq * 16 + tid];
      const float hi = (gg >= 0.f) ? gg * qmx : gg * qmn;
      const float lo = (gg >= 0.f) ? gg * qmn : gg * qmx;
      bmax = fmaxf(bmax, hi);
      bmin = fminf(bmin, lo);
    }
    sBmax[tid] = bmax; sBmin[tid] = bmin;
  }
  __syncthreads();

  const int j  = blockIdx.y * 256 + tid;
  const int jm = j >> 4, jq = j & 15;
  const float alpha = SCALEF * k[b * Msz + h * 128 + jm];
  const float mxv = (alpha > 0.f) ? alpha * sBmax[jq] : alpha * sBmin[jq];
  float den = 0.f;
  for (int iq = 0; iq < 16; ++iq) {
    const float ag = alpha * sG[iq * 16 + jq];
    for (int im = 0; im < 128; ++im)
      den += __expf(ag * sQ[im] - mxv);
  }
  ws[WS_MX + bh * 2048 + j] = mxv;
  ws[WS_VD + bh * 2048 + j] = v[b * Msz + h * 128 + jm] / den;  // den >= 1
}

// ---------------------------------------------------------------------------
// Kernel C (pass 2): per row i=(im,iq):
//   U[i] = sum_jq Gvf[jq,iq] * sum_jm exp(cK[jm]*Qm[im]*G[iq,jq] - mx[j]) * vd[j]
// then out'(im) = sum over the 16 i sharing im (t = iq).
// grid (64, 8) x 256: one thread per row i; smx/svd reads are wave-uniform
// (broadcast, conflict-free).
// ---------------------------------------------------------------------------
__global__ __launch_bounds__(256)
void pass2_kernel(const float* __restrict__ q, const float* __restrict__ k,
                  float* __restrict__ ws) {
  __shared__ float sQ[128], scK[128];
  __shared__ float sG[256], sGvf[256];
  __shared__ float smx[2048], svd[2048];
  __shared__ float sU[256];
  const int bh = blockIdx.x;
  const int b = bh >> 4, h = bh & 15;
  const int tid = threadIdx.x;

  if (tid < 128) {
    sQ[tid]  = q[b * Msz + h * 128 + tid];
    scK[tid] = SCALEF * k[b * Msz + h * 128 + tid];
  }
  sG[tid]   = ws[WS_G + tid];
  sGvf[tid] = ws[WS_GVF + tid];
  for (int t = tid; t < 2048; t += 256) {
    smx[t] = ws[WS_MX + bh * 2048 + t];
    svd[t] = ws[WS_VD + bh * 2048 + t];
  }
  __syncthreads();

  const int i  = blockIdx.y * 256 + tid;
  const int im = i >> 4, iq = i & 15;
  const float qi = sQ[im];
  float U = 0.f;
  for (int jq = 0; jq < 16; ++jq) {
    const float p = qi * sG[iq * 16 + jq];
    float acc = 0.f;
    for (int jm = 0; jm < 128; ++jm) {
      const int j = jm * 16 + jq;
      acc += __expf(scK[jm] * p - smx[j]) * svd[j];
    }
    U += sGvf[jq * 16 + iq] * acc;
  }
  sU[tid] = U;
  __syncthreads();
  if (tid < 16) {   // sum the 16 t-values of one im
    float s = 0.f;
    for (int t = 0; t < 16; ++t) s += sU[tid * 16 + t];
    ws[WS_AO + b * Msz + h * 128 + blockIdx.y * 16 + tid] = s;
  }
}

// ---------------------------------------------------------------------------
// Kernel D: residual add + LayerNorm over M=2048.  grid 4 x 256, 8 elems/thr.
// ---------------------------------------------------------------------------
__global__ __launch_bounds__(256)
void ln_kernel(const float* __restrict__ q, const float* __restrict__ lnw,
               const float* __restrict__ lnb, const float* __restrict__ ws,
               float* __restrict__ out) {
  __shared__ float red[256];
  __shared__ float stat[2];
  const int b = blockIdx.x, tid = threadIdx.x;
  float y[8];
  float s = 0.f;
#pragma unroll
  for (int c = 0; c < 8; ++c) {
    const int m = tid + c * 256;
    y[c] = q[b * Msz + m] + ws[WS_AO + b * Msz + m];
    s += y[c];
  }
  red[tid] = s; __syncthreads();
  for (int off = 128; off > 0; off >>= 1) {
    if (tid < off) red[tid] += red[tid + off];
    __syncthreads();
  }
  if (tid == 0) stat[0] = red[0] * (1.f / 2048.f);
  __syncthreads();
  const float mu = stat[0];
  s = 0.f;
#pragma unroll
  for (int c = 0; c < 8; ++c) { const float d = y[c] - mu; s += d * d; }
  red[tid] = s; __syncthreads();
  for (int off = 128; off > 0; off >>= 1) {
    if (tid < off) red[tid] += red[tid + off];
    __syncthreads();
  }
  if (tid == 0) stat[1] = red[0] * (1.f / 2048.f);
  __syncthreads();
  const float rinv = rsqrtf(stat[1] + LN_EPS);
#pragma unroll
  for (int c = 0; c < 8; ++c) {
    const int m = tid + c * 256;
    out[b * Msz + m] = (y[c] - mu) * rinv * lnw[m] + lnb[m];
  }
}

extern "C" void kernel_launch(void* const* d_in, const int* in_sizes, int n_in,
                              void* d_out, int out_size, void* d_ws, size_t ws_size,
                              hipStream_t stream) {
  (void)in_sizes; (void)n_in; (void)out_size; (void)ws_size;
  const float* q   = (const float*)d_in[0];
  const float* k   = (const float*)d_in[1];
  const float* v   = (const float*)d_in[2];
  const float* wq  = (const float*)d_in[3];
  const float* wk  = (const float*)d_in[4];
  const float* wv  = (const float*)d_in[5];
  const float* wf  = (const float*)d_in[6];
  const float* lnw = (const float*)d_in[7];
  const float* lnb = (const float*)d_in[8];
  float* ws  = (float*)d_ws;
  float* out = (float*)d_out;

  gram_wmma_kernel<<<1, 32, 0, stream>>>(wq, wk, wv, wf, ws);
  pass1_kernel<<<dim3(64, 8), 256, 0, stream>>>(q, k, v, ws);
  pass2_kernel<<<dim3(64, 8), 256, 0, stream>>>(q, k, ws);
  ln_kernel<<<4, 256, 0, stream>>>(q, lnw, lnb, ws, out);
}